// ModelNew_80908593922806
// MI455X (gfx1250) — compile-verified
//
#include <hip/hip_runtime.h>
#include <math.h>

typedef __attribute__((ext_vector_type(2))) float v2f;
typedef __attribute__((ext_vector_type(8))) float v8f;

#define IN_C  16
#define OUT_C 64
#define H     256
#define W     256
#define OH    254
#define OW    254
#define KTOT  144
#define NIMG  64

// Byte offset into x for the k-th reduction element (k = ic*9 + kh*3 + kw).
constexpr int koffB(int k) {
    return ((k / 9) * (H * W) + ((k % 9) / 3) * W + (k % 9) % 3) * 4;
}

__global__ __launch_bounds__(256)
void conv_min_tanh_wmma(const float* __restrict__ x,
                        const float* __restrict__ wgt,
                        const float* __restrict__ bias,
                        float* __restrict__ out)
{
    // Pair-interleaved weights: B2[p][n][e], p = k>>1, e = k&1.
    // Fragment {B[k][n], B[k+1][n]} is one contiguous 8 B -> ds_load_b64.
    __shared__ float ldsB2[(KTOT / 2) * OUT_C * 2];   // 36 KB

    const int tid = threadIdx.x;
    for (int idx = tid; idx < KTOT * OUT_C; idx += 256) {
        int n = idx / KTOT;          // out channel (coalesced read of OIHW flat)
        int k = idx % KTOT;
        ldsB2[(k >> 1) * (OUT_C * 2) + n * 2 + (k & 1)] = wgt[idx];
    }
    __syncthreads();

    const int lane = tid & 31;
    const int wave = tid >> 5;
    const int half = lane >> 4;      // 0: lanes 0-15, 1: lanes 16-31
    const int l16  = lane & 15;

    const int n_img = blockIdx.y;
    const int oh    = blockIdx.x;
    const int ow0   = wave * 32;     // two 16-pixel M tiles per wave

    // Per-lane clamped 32-bit element base. LIMIT == largest base of any
    // in-range output pixel, so clamping never disturbs a valid pixel; rows
    // with ow >= 254 get garbage and are discarded at store. Guarantees
    // base + max_koff <= xTotal-1 (no OOB reads).
    const unsigned LIMIT = (unsigned)NIMG * IN_C * H * W - 1u
                         - (unsigned)((IN_C - 1) * H * W + 2 * W + 2);
    unsigned pA = (unsigned)n_img * (IN_C * H * W) + (unsigned)oh * W + (ow0 + l16);
    unsigned pB = pA + 16u;
    if (pA > LIMIT) pA = LIMIT;
    if (pB > LIMIT) pB = LIMIT;
    const unsigned pAb = pA << 2;    // byte offsets (< 2^28, fits 32-bit)
    const unsigned pBb = pB << 2;
    const char* xb = (const char*)x;

    // Per-lane LDS base for B fragments; per-step/tile deltas are immediates.
    const char* Bb = (const char*)ldsB2 + (unsigned)half * (OUT_C * 8)
                   + (unsigned)l16 * 8;

    v8f c00 = {}, c01 = {}, c02 = {}, c03 = {};   // M tile 0 x 4 N tiles
    v8f c10 = {}, c11 = {}, c12 = {}, c13 = {};   // M tile 1 x 4 N tiles

    const float bi0 = bias[l16];
    const float bi1 = bias[16 + l16];
    const float bi2 = bias[32 + l16];
    const float bi3 = bias[48 + l16];

    #pragma unroll
    for (int k0 = 0; k0 < KTOT; k0 += 4) {
        // A layout (16x4 f32): M = l16; VGPR0 K = k0+2*half, VGPR1 K = k0+1+2*half.
        const int o0 = half ? koffB(k0 + 2) : koffB(k0);
        const int o1 = half ? koffB(k0 + 3) : koffB(k0 + 1);

        v2f a0, a1;
        a0[0] = *(const float*)(xb + (pAb + (unsigned)o0));
        a0[1] = *(const float*)(xb + (pAb + (unsigned)o1));
        a1[0] = *(const float*)(xb + (pBb + (unsigned)o0));
        a1[1] = *(const float*)(xb + (pBb + (unsigned)o1));

        // B fragments: one ds_load_b64 per N tile, shared address VGPR.
        const char* bp = Bb + (k0 >> 1) * (OUT_C * 8);
        v2f b0 = *(const v2f*)(bp + 0);
        v2f b1 = *(const v2f*)(bp + 128);
        v2f b2 = *(const v2f*)(bp + 256);
        v2f b3 = *(const v2f*)(bp + 384);

        c00 = __builtin_amdgcn_wmma_f32_16x16x4_f32(false, a0, false, b0, (short)0, c00, false, false);
        c10 = __builtin_amdgcn_wmma_f32_16x16x4_f32(false, a1, false, b0, (short)0, c10, false, false);
        c01 = __builtin_amdgcn_wmma_f32_16x16x4_f32(false, a0, false, b1, (short)0, c01, false, false);
        c11 = __builtin_amdgcn_wmma_f32_16x16x4_f32(false, a1, false, b1, (short)0, c11, false, false);
        c02 = __builtin_amdgcn_wmma_f32_16x16x4_f32(false, a0, false, b2, (short)0, c02, false, false);
        c12 = __builtin_amdgcn_wmma_f32_16x16x4_f32(false, a1, false, b2, (short)0, c12, false, false);
        c03 = __builtin_amdgcn_wmma_f32_16x16x4_f32(false, a0, false, b3, (short)0, c03, false, false);
        c13 = __builtin_amdgcn_wmma_f32_16x16x4_f32(false, a1, false, b3, (short)0, c13, false, false);
    }

    // Epilogue (divergence only after all WMMAs): C layout puts (M = r + 8*half,
    // N = l16 + 16*tile) in VGPR r. Bias-add, min over 4 N tiles in-lane, then
    // xor-min across the 16-lane half, tanh(tanh(.)), predicated store.
    const long long outRow = ((long long)n_img * OH + oh) * OW;

    #pragma unroll
    for (int r = 0; r < 8; ++r) {
        float v0 = fminf(fminf(c00[r] + bi0, c01[r] + bi1),
                         fminf(c02[r] + bi2, c03[r] + bi3));
        float v1 = fminf(fminf(c10[r] + bi0, c11[r] + bi1),
                         fminf(c12[r] + bi2, c13[r] + bi3));
        v0 = fminf(v0, __shfl_xor(v0, 1, 32));
        v1 = fminf(v1, __shfl_xor(v1, 1, 32));
        v0 = fminf(v0, __shfl_xor(v0, 2, 32));
        v1 = fminf(v1, __shfl_xor(v1, 2, 32));
        v0 = fminf(v0, __shfl_xor(v0, 4, 32));
        v1 = fminf(v1, __shfl_xor(v1, 4, 32));
        v0 = fminf(v0, __shfl_xor(v0, 8, 32));
        v1 = fminf(v1, __shfl_xor(v1, 8, 32));
        if (l16 == 0) {
            const int m   = r + 8 * half;
            const int ow  = ow0 + m;
            const int ow2 = ow + 16;
            if (ow < OW)  out[outRow + ow]  = tanhf(tanhf(v0));
            if (ow2 < OW) out[outRow + ow2] = tanhf(tanhf(v1));
        }
    }
}

extern "C" void kernel_launch(void* const* d_in, const int* in_sizes, int n_in,
                              void* d_out, int out_size, void* d_ws, size_t ws_size,
                              hipStream_t stream)
{
    const float* x    = (const float*)d_in[0];   // (64,16,256,256) f32
    const float* wgt  = (const float*)d_in[1];   // (64,16,3,3) f32
    const float* bias = (const float*)d_in[2];   // (64,) f32
    float* out        = (float*)d_out;           // (64,1,254,254) f32

    dim3 grid(OH, NIMG);   // one block per output row per image
    dim3 block(256);       // 8 wave32, each covering 32 pixels
    conv_min_tanh_wmma<<<grid, block, 0, stream>>>(x, wgt, bias, out);
}